// AnomalyBlock_46480136077816
// MI455X (gfx1250) — compile-verified
//
#include <hip/hip_runtime.h>
#include <hip/hip_bf16.h>

#define B_  8
#define L_  1024
#define DM  256
#define H_  8
#define DK  32

typedef __attribute__((ext_vector_type(2))) float v2f;
typedef __attribute__((ext_vector_type(8))) float v8f;
typedef unsigned int u32x4 __attribute__((ext_vector_type(4)));
typedef int i32x4 __attribute__((ext_vector_type(4)));
typedef int i32x8 __attribute__((ext_vector_type(8)));

#define AS1 __attribute__((address_space(1)))
#define AS3 __attribute__((address_space(3)))

#if defined(__has_builtin)
#if __has_builtin(__builtin_amdgcn_global_load_async_to_lds_b128)
#define HAVE_ASYNC_LDS 1
#endif
#if __has_builtin(__builtin_amdgcn_tensor_load_to_lds)
#define HAVE_TDM 1
#endif
#endif

__device__ __forceinline__ void wait_async0() {
#if defined(__has_builtin) && __has_builtin(__builtin_amdgcn_s_wait_asynccnt)
  __builtin_amdgcn_s_wait_asynccnt(0);
#else
  asm volatile("s_wait_asynccnt 0x0" ::: "memory");
#endif
}

__device__ __forceinline__ v8f wmma4(v2f a, v2f b, v8f c) {
  // D = A(16x4,f32) x B(4x16,f32) + C(16x16,f32)
  return __builtin_amdgcn_wmma_f32_16x16x4_f32(false, a, false, b, (short)0, c, false, false);
}

#ifdef HAVE_ASYNC_LDS
// b128 async copy: 16 bytes per lane, global -> LDS, tracked by ASYNCcnt.
__device__ __forceinline__ void async_cp16(const float* gsrc, float* lds_dst) {
  __builtin_amdgcn_global_load_async_to_lds_b128(
      (AS1 i32x4*)gsrc, (AS3 i32x4*)lds_dst, 0, 0);
}
#endif

#ifdef HAVE_TDM
// TDM: DMA a [1024 x 32] fp32 tile (global row stride 256 floats) into LDS,
// inserting 1 DWORD of padding after every 32 DWORDs (row stride 33 in LDS).
__device__ __forceinline__ void tdm_load_tile(const float* gsrc, void* lds_dst) {
  unsigned long long ga = (unsigned long long)(uintptr_t)gsrc;
  unsigned int la = (unsigned int)(uintptr_t)lds_dst;   // low 32 bits = LDS offset
  u32x4 g0;
  g0[0] = 1u;                                            // count=1, user mode
  g0[1] = la;                                            // lds_addr
  g0[2] = (unsigned int)(ga & 0xffffffffu);              // global_addr[31:0]
  g0[3] = (unsigned int)((ga >> 32) & 0x01ffffffu)       // global_addr[56:32]
        | (2u << 30);                                    // type=2 ("image")
  i32x8 g1;
  g1[0] = (2 << 16)        // data_size = 4 bytes
        | (1 << 20)        // pad_enable
        | (4 << 22);       // pad_interval: pad after 32 DWORDs
                           // pad_amount = 0 -> 1 DWORD
  g1[1] = (DK << 16);      // tensor_dim0 = 32 (bits 79:48, low half)
  g1[2] = (L_ << 16);      // tensor_dim0 hi=0 ; tensor_dim1 = 1024 (low half)
  g1[3] = (DK << 16);      // tensor_dim1 hi=0 ; tile_dim0 = 32
  g1[4] = L_;              // tile_dim1 = 1024 ; tile_dim2 = 0
  g1[5] = DM;              // tensor_dim0_stride = 256 elements
  g1[6] = 0;               // stride hi / dim1_stride lo
  g1[7] = 0;
  i32x4 z4 = {0, 0, 0, 0};
#if __clang_major__ >= 23
  i32x8 z8 = {0, 0, 0, 0, 0, 0, 0, 0};
  __builtin_amdgcn_tensor_load_to_lds(g0, g1, z4, z4, z8, 0);
#else
  __builtin_amdgcn_tensor_load_to_lds(g0, g1, z4, z4, 0);
#endif
}
#endif

// ---------------------------------------------------------------------------
// Generic C[M,256] = A[M,256] @ W[256,256] + bias, fp32 WMMA.
// Block: 128 threads (4 waves). Block tile 64x64; wave computes 16x64.
// Tile staging uses the async global->LDS path (ASYNCcnt) when available.
// ---------------------------------------------------------------------------
__global__ __launch_bounds__(128)
void gemm256_bias(const float* __restrict__ A, const float* __restrict__ W,
                  const float* __restrict__ bias, float* __restrict__ C)
{
  __shared__ float As[64][20];  // 64 rows x 16 K (pad->20)
  __shared__ float Bs[16][68];  // 16 K x 64 N (pad)

  const int tid  = threadIdx.x;
  const int wave = tid >> 5;      // 0..3
  const int lane = tid & 31;
  const int half = lane >> 4;
  const int m16  = lane & 15;
  const int rowBase = blockIdx.x * 64;
  const int colBase = blockIdx.y * 64;

  v8f acc[4];
  #pragma unroll
  for (int n = 0; n < 4; ++n)
    #pragma unroll
    for (int i = 0; i < 8; ++i) acc[n][i] = 0.0f;

  for (int k0 = 0; k0 < DM; k0 += 16) {
#ifdef HAVE_ASYNC_LDS
    #pragma unroll
    for (int it = 0; it < 2; ++it) {
      int idx = tid + it * 128;
      int r   = idx >> 2;
      int c4  = (idx & 3) * 4;
      async_cp16(A + (size_t)(rowBase + r) * DM + k0 + c4, &As[r][c4]);
    }
    #pragma unroll
    for (int it = 0; it < 2; ++it) {
      int idx = tid + it * 128;
      int kk  = idx >> 4;
      int n4  = (idx & 15) * 4;
      async_cp16(W + (size_t)(k0 + kk) * DM + colBase + n4, &Bs[kk][n4]);
    }
    wait_async0();
#else
    #pragma unroll
    for (int it = 0; it < 2; ++it) {
      int idx = tid + it * 128;
      int r   = idx >> 2;
      int c4  = (idx & 3) * 4;
      float4 val = *(const float4*)(A + (size_t)(rowBase + r) * DM + k0 + c4);
      As[r][c4 + 0] = val.x; As[r][c4 + 1] = val.y;
      As[r][c4 + 2] = val.z; As[r][c4 + 3] = val.w;
    }
    #pragma unroll
    for (int it = 0; it < 2; ++it) {
      int idx = tid + it * 128;
      int kk  = idx >> 4;
      int n4  = (idx & 15) * 4;
      float4 val = *(const float4*)(W + (size_t)(k0 + kk) * DM + colBase + n4);
      Bs[kk][n4 + 0] = val.x; Bs[kk][n4 + 1] = val.y;
      Bs[kk][n4 + 2] = val.z; Bs[kk][n4 + 3] = val.w;
    }
#endif
    if (k0 + 16 < DM) {   // prefetch next slab into L2
      __builtin_prefetch(A + (size_t)(rowBase + (tid & 63)) * DM + k0 + 16, 0, 1);
      __builtin_prefetch(W + (size_t)(k0 + 16 + (tid & 15)) * DM + colBase, 0, 1);
    }
    __syncthreads();

    #pragma unroll
    for (int kk = 0; kk < 16; kk += 4) {
      v2f a;
      a[0] = As[wave * 16 + m16][kk + 2 * half];
      a[1] = As[wave * 16 + m16][kk + 2 * half + 1];
      #pragma unroll
      for (int n = 0; n < 4; ++n) {
        v2f b;
        b[0] = Bs[kk + 2 * half][n * 16 + m16];
        b[1] = Bs[kk + 2 * half + 1][n * 16 + m16];
        acc[n] = wmma4(a, b, acc[n]);
      }
    }
    __syncthreads();
  }

  #pragma unroll
  for (int n = 0; n < 4; ++n) {
    int col = colBase + n * 16 + m16;
    float bv = bias[col];
    #pragma unroll
    for (int i = 0; i < 8; ++i) {
      int rm = i + 8 * half;
      C[(size_t)(rowBase + wave * 16 + rm) * DM + col] = acc[n][i] + bv;
    }
  }
}

// ---------------------------------------------------------------------------
// Fused attention for one (b,h) and a 16-row query tile.
// K (then V) of one (b,h) staged via TDM into 132KB of LDS; score tile kept
// in LDS across score->softmax->attn@V. Block: 256 threads (8 waves).
// ---------------------------------------------------------------------------
__global__ __launch_bounds__(256)
void attention16(const float* __restrict__ q, const float* __restrict__ k,
                 const float* __restrict__ v, float* __restrict__ series,
                 float* __restrict__ weighted)
{
  __shared__ float KV[L_][DK + 1];     // 132 KB; holds K, then reused for V
  __shared__ float S[16][L_ + 4];      // 66 KB score / attention tile
  __shared__ float Qs[16][DK + 1];
  __shared__ float red[16][17];
  __shared__ float rowstat[16];
  __shared__ float wacc[16][DK];

  const int tid  = threadIdx.x;
  const int wave = tid >> 5;   // 0..7
  const int lane = tid & 31;
  const int half = lane >> 4;
  const int m16  = lane & 15;
  const int bh   = blockIdx.y;        // b*H + h
  const int b    = bh >> 3;
  const int h    = bh & 7;
  const int l0   = blockIdx.x * 16;
  const float scale = 0.17677669529663687f;  // 32^-0.5

  // ---- Stage K[b,h] : 1024x32 strided tensor -> padded LDS ----
#ifdef HAVE_TDM
  if (wave == 0)
    tdm_load_tile(k + (size_t)(b * L_) * DM + h * DK, &KV[0][0]);
#else
  for (int it = 0; it < 32; ++it) {
    int idx = tid + it * 256;
    int r   = idx >> 3;
    int c4  = (idx & 7) * 4;
    float4 val = *(const float4*)(k + (size_t)(b * L_ + r) * DM + h * DK + c4);
    KV[r][c4 + 0] = val.x; KV[r][c4 + 1] = val.y;
    KV[r][c4 + 2] = val.z; KV[r][c4 + 3] = val.w;
  }
#endif
  // ---- Stage Q tile (pre-scaled) ----
  if (tid < 128) {
    int r  = tid >> 3;
    int c4 = (tid & 7) * 4;
    float4 val = *(const float4*)(q + (size_t)(b * L_ + l0 + r) * DM + h * DK + c4);
    Qs[r][c4 + 0] = val.x * scale; Qs[r][c4 + 1] = val.y * scale;
    Qs[r][c4 + 2] = val.z * scale; Qs[r][c4 + 3] = val.w * scale;
  }
#ifdef HAVE_TDM
  if (wave == 0) __builtin_amdgcn_s_wait_tensorcnt(0);
#endif
  __syncthreads();

  // ---- Phase 1: scores = Q_tile @ K^T, per-wave 128-column strip ----
  const int colW = wave * 128;
  for (int ct = 0; ct < 8; ++ct) {
    v8f acc;
    #pragma unroll
    for (int i = 0; i < 8; ++i) acc[i] = 0.0f;
    int nb = colW + ct * 16;
    #pragma unroll
    for (int kd = 0; kd < DK; kd += 4) {
      v2f a;
      a[0] = Qs[m16][kd + 2 * half];
      a[1] = Qs[m16][kd + 2 * half + 1];
      v2f bb;  // B[k][n] = K[n][k]
      bb[0] = KV[nb + m16][kd + 2 * half];
      bb[1] = KV[nb + m16][kd + 2 * half + 1];
      acc = wmma4(a, bb, acc);
    }
    #pragma unroll
    for (int i = 0; i < 8; ++i)
      S[i + 8 * half][nb + m16] = acc[i];
  }
  __syncthreads();

  // ---- Softmax over each full row of 1024 (16 threads per row) ----
  {
    int r = tid >> 4, c = tid & 15;
    float mx = -3.4e38f;
    for (int s = c; s < L_; s += 16) mx = fmaxf(mx, S[r][s]);
    red[r][c] = mx;
  }
  __syncthreads();
  if (tid < 16) {
    float mx = red[tid][0];
    #pragma unroll
    for (int c = 1; c < 16; ++c) mx = fmaxf(mx, red[tid][c]);
    rowstat[tid] = mx;
  }
  __syncthreads();
  {
    int r = tid >> 4, c = tid & 15;
    float mx = rowstat[r];
    float sum = 0.0f;
    for (int s = c; s < L_; s += 16) {
      float e = __expf(S[r][s] - mx);
      S[r][s] = e;
      sum += e;
    }
    red[r][c] = sum;
  }
  __syncthreads();
  if (tid < 16) {
    float sum = 0.0f;
    #pragma unroll
    for (int c = 0; c < 16; ++c) sum += red[tid][c];
    rowstat[tid] = 1.0f / sum;
  }
  __syncthreads();
  {
    int r = tid >> 4, c = tid & 15;
    float inv = rowstat[r];
    for (int s = c; s < L_; s += 16) S[r][s] *= inv;
  }
  __syncthreads();

  // ---- Stage V[b,h] into same LDS region (phase-1 reads are done) ----
#ifdef HAVE_TDM
  if (wave == 0)
    tdm_load_tile(v + (size_t)(b * L_) * DM + h * DK, &KV[0][0]);
#endif

  // ---- Write series_attn tile once (coalesced float4) ----
  for (int it = 0; it < 16; ++it) {
    int idx = tid + it * 256;          // 0..4095 float4 slots
    int r   = idx >> 8;
    int s4  = (idx & 255) * 4;
    float4 o;
    o.x = S[r][s4 + 0]; o.y = S[r][s4 + 1];
    o.z = S[r][s4 + 2]; o.w = S[r][s4 + 3];
    *(float4*)(series + ((size_t)bh * L_ + l0 + r) * L_ + s4) = o;
  }

#ifndef HAVE_TDM
  for (int it = 0; it < 32; ++it) {
    int idx = tid + it * 256;
    int r   = idx >> 3;
    int c4  = (idx & 7) * 4;
    float4 val = *(const float4*)(v + (size_t)(b * L_ + r) * DM + h * DK + c4);
    KV[r][c4 + 0] = val.x; KV[r][c4 + 1] = val.y;
    KV[r][c4 + 2] = val.z; KV[r][c4 + 3] = val.w;
  }
#endif
  #pragma unroll
  for (int it = 0; it < 2; ++it) {
    int idx = tid + it * 256;
    wacc[idx >> 5][idx & 31] = 0.0f;
  }
#ifdef HAVE_TDM
  if (wave == 0) __builtin_amdgcn_s_wait_tensorcnt(0);
#endif
  __syncthreads();

  // ---- Phase 2: weighted = attn @ V ; K split over 8 waves ----
  const int kbase = wave * 128;
  #pragma unroll 1
  for (int nt = 0; nt < 2; ++nt) {
    v8f acc;
    #pragma unroll
    for (int i = 0; i < 8; ++i) acc[i] = 0.0f;
    #pragma unroll 8
    for (int kk = 0; kk < 128; kk += 4) {
      v2f a;
      a[0] = S[m16][kbase + kk + 2 * half];
      a[1] = S[m16][kbase + kk + 2 * half + 1];
      v2f bb;
      bb[0] = KV[kbase + kk + 2 * half][nt * 16 + m16];
      bb[1] = KV[kbase + kk + 2 * half + 1][nt * 16 + m16];
      acc = wmma4(a, bb, acc);
    }
    #pragma unroll
    for (int i = 0; i < 8; ++i)
      atomicAdd(&wacc[i + 8 * half][nt * 16 + m16], acc[i]);
  }
  __syncthreads();

  // ---- Write weighted tile [16 x 32] into [B*L, 256] head-major buffer ----
  #pragma unroll
  for (int it = 0; it < 2; ++it) {
    int idx = tid + it * 256;
    int r = idx >> 5, c = idx & 31;
    weighted[(size_t)(b * L_ + l0 + r) * DM + h * DK + c] = wacc[r][c];
  }
}

// ---------------------------------------------------------------------------
// Prior tables: expT[h][d] = exp(-d^2/(2 sigma_h^2)), rowsum[h][l].
// ---------------------------------------------------------------------------
__global__ __launch_bounds__(1024)
void prior_tables(const float* __restrict__ sigma_in, float* __restrict__ expT,
                  float* __restrict__ rowsum)
{
  __shared__ float e[L_];
  const int h = blockIdx.x;
  const int t = threadIdx.x;
  float s = fabsf(sigma_in[h]) + 1e-6f;
  float inv2s2 = 1.0f / (2.0f * s * s);
  float d = (float)t;
  float val = __expf(-d * d * inv2s2);
  e[t] = val;
  expT[h * L_ + t] = val;
  __syncthreads();
  float sum = 0.0f;
  for (int sc = 0; sc < L_; ++sc) {
    int dd = t - sc; dd = dd < 0 ? -dd : dd;
    sum += e[dd];
  }
  rowsum[h * L_ + t] = sum;
}

// ---------------------------------------------------------------------------
// Stream 256 MB prior_attn at HBM rate; tables are L2-resident.
// ---------------------------------------------------------------------------
__global__ __launch_bounds__(256)
void prior_write(const float* __restrict__ expT, const float* __restrict__ rowsum,
                 float* __restrict__ prior)
{
  size_t gid = (size_t)blockIdx.x * 256 + threadIdx.x;
  size_t s4  = gid * 4;                       // flat index into [B,H,L,L]
  int s = (int)(s4 & (L_ - 1));
  int l = (int)((s4 >> 10) & (L_ - 1));
  int h = (int)((s4 >> 20) & (H_ - 1));
  float inv = 1.0f / (rowsum[h * L_ + l] + 1e-8f);
  const float* et = expT + h * L_;
  float4 o;
  int d0 = l - s;       d0 = d0 < 0 ? -d0 : d0;
  int d1 = l - (s + 1); d1 = d1 < 0 ? -d1 : d1;
  int d2 = l - (s + 2); d2 = d2 < 0 ? -d2 : d2;
  int d3 = l - (s + 3); d3 = d3 < 0 ? -d3 : d3;
  o.x = et[d0] * inv; o.y = et[d1] * inv;
  o.z = et[d2] * inv; o.w = et[d3] * inv;
  *(float4*)(prior + s4) = o;
}

// ---------------------------------------------------------------------------
extern "C" void kernel_launch(void* const* d_in, const int* in_sizes, int n_in,
                              void* d_out, int out_size, void* d_ws, size_t ws_size,
                              hipStream_t stream) {
  const float* x     = (const float*)d_in[0];
  const float* Wq    = (const float*)d_in[1];
  const float* bq    = (const float*)d_in[2];
  const float* Wk    = (const float*)d_in[3];
  const float* bk    = (const float*)d_in[4];
  const float* Wv    = (const float*)d_in[5];
  const float* bv    = (const float*)d_in[6];
  const float* Wo    = (const float*)d_in[7];
  const float* bo    = (const float*)d_in[8];
  const float* sigma = (const float*)d_in[9];

  float* out    = (float*)d_out;
  float* series = out + (size_t)B_ * L_ * DM;                 // [B,H,L,L]
  float* prior  = series + (size_t)B_ * H_ * L_ * L_;         // [B,H,L,L]

  const size_t MT = (size_t)B_ * L_;  // 8192 tokens
  float* ws   = (float*)d_ws;
  float* qb   = ws;                   // [MT, 256]
  float* kb   = qb + MT * DM;
  float* vb   = kb + MT * DM;
  float* wt   = vb + MT * DM;         // weighted, [MT, 256]
  float* expT = wt + MT * DM;         // [H, L]
  float* rows = expT + (size_t)H_ * L_;

  dim3 gg((unsigned)(MT / 64), DM / 64);
  gemm256_bias<<<gg, 128, 0, stream>>>(x, Wq, bq, qb);
  gemm256_bias<<<gg, 128, 0, stream>>>(x, Wk, bk, kb);
  gemm256_bias<<<gg, 128, 0, stream>>>(x, Wv, bv, vb);

  prior_tables<<<H_, L_, 0, stream>>>(sigma, expT, rows);
  prior_write<<<(unsigned)(((size_t)B_ * H_ * L_ * L_) / 4 / 256), 256, 0, stream>>>(expT, rows, prior);

  attention16<<<dim3(L_ / 16, B_ * H_), 256, 0, stream>>>(qb, kb, vb, series, wt);

  gemm256_bias<<<gg, 128, 0, stream>>>(wt, Wo, bo, out);
}